// YOLOLoss_60395830117010
// MI455X (gfx1250) — compile-verified
//
#include <hip/hip_runtime.h>
#include <hip/hip_bf16.h>

// ---- problem constants (from reference) ----
#define LAMBDA_BOX   5.0f
#define LAMBDA_OBJ   1.0f
#define LAMBDA_NOOBJ 0.5f
#define LAMBDA_CLS   1.0f

static constexpr int BB  = 16;   // batch
static constexpr int AA  = 3;    // anchors
static constexpr int GH  = 80;
static constexpr int GW  = 80;
static constexpr int DD  = 85;   // 5 + C
static constexpr int CC  = 80;   // classes
static constexpr int MM  = 50;   // targets per image
static constexpr int NCELLS = BB * AA * GH * GW;   // 307200
static constexpr int NB2 = 640;                    // blocks for main kernel
static constexpr int NT2 = 256;                    // threads per block (8 waves)

typedef __attribute__((ext_vector_type(2))) float v2f;
typedef __attribute__((ext_vector_type(8))) float v8f;

// ---------------------------------------------------------------------------
// Wave-level sum using V_WMMA_F32_16X16X4_F32:
//   A = ones(16x4), B[4x16] holds the 32 per-lane partials (b0 lanes 0..31 fill
//   B[0][*] and B[2][*]; b1 = 0). D[m][n] = p_n + p_{n+16} for every row m.
//   Each lane then holds 8 identical copies of its column sum; 4 xor-shuffles
//   across a 16-lane half finish the reduction. Result = wave sum in all lanes.
// EXEC must be all ones: call only from fully-active code.
// ---------------------------------------------------------------------------
__device__ __forceinline__ float wave_sum_wmma(float p) {
    v2f a; a[0] = 1.0f; a[1] = 1.0f;
    v2f b; b[0] = p;    b[1] = 0.0f;
    v8f c = {};
    c = __builtin_amdgcn_wmma_f32_16x16x4_f32(false, a, false, b,
                                              (short)0, c, false, false);
    float s = c[0] + c[1] + c[2] + c[3] + c[4] + c[5] + c[6] + c[7]; // 8*(p_n+p_{n+16})
    s += __shfl_xor(s, 1, 32);
    s += __shfl_xor(s, 2, 32);
    s += __shfl_xor(s, 4, 32);
    s += __shfl_xor(s, 8, 32);   // halves are duplicates; each half sums itself
    return s * 0.125f;           // = full 32-lane sum
}

// ---------------------------------------------------------------------------
// K0: zero the cell->target map
// ---------------------------------------------------------------------------
__global__ void k_zero_map(int* __restrict__ map) {
    int i = blockIdx.x * blockDim.x + threadIdx.x;
    if (i < NCELLS) map[i] = 0;
}

// ---------------------------------------------------------------------------
// K1: assign each target to its best-anchor cell. Duplicates resolved with
// atomicMax on (m+1): order-independent -> deterministic.
// ---------------------------------------------------------------------------
__global__ void k_assign(const float* __restrict__ tgt_box,
                         const int*   __restrict__ tgt_cls,
                         const float* __restrict__ anchors,
                         int* __restrict__ map) {
    int t = blockIdx.x * blockDim.x + threadIdx.x;
    if (t >= BB * MM) return;
    int b = t / MM, m = t - b * MM;
    const float* tb = tgt_box + (size_t)(b * MM + m) * 4;
    float x = tb[0], y = tb[1], w = tb[2], h = tb[3];
    float gx = x * GW, gy = y * GH;
    float gif = floorf(gx), gjf = floorf(gy);
    int gi = (int)gif, gj = (int)gjf;
    gi = min(max(gi, 0), GW - 1);
    gj = min(max(gj, 0), GH - 1);
    float bx = gx - gif, by = gy - gjf;
    float bw = w * GW,  bh = h * GH;

    float bx1 = bx - 0.5f * bw, by1 = by - 0.5f * bh;
    float bx2 = bx + 0.5f * bw, by2 = by + 0.5f * bh;
    float area1 = fmaxf(bw, 0.0f) * fmaxf(bh, 0.0f);

    int best = 0; float best_iou = -1.0f;
    #pragma unroll
    for (int a = 0; a < AA; ++a) {
        float aw = anchors[a * 2 + 0], ah = anchors[a * 2 + 1];
        float x1 = fmaxf(bx1, -0.5f * aw);
        float y1 = fmaxf(by1, -0.5f * ah);
        float x2 = fminf(bx2,  0.5f * aw);
        float y2 = fminf(by2,  0.5f * ah);
        float inter = fmaxf(x2 - x1, 0.0f) * fmaxf(y2 - y1, 0.0f);
        float iou = inter / (area1 + aw * ah - inter + 1e-7f);
        if (iou > best_iou) { best_iou = iou; best = a; }  // first wins ties
    }
    int cell = ((b * AA + best) * GH + gj) * GW + gi;
    atomicMax(&map[cell], m + 1);
}

// ---------------------------------------------------------------------------
// K2: main streaming kernel over all cells. Miss path reads only the obj
// channel (stride-85 loads; ~39 MB of line traffic). Hit path (<=800 cells)
// reads the full 85-float cell and recomputes targets. Block partial written
// to d_ws; wave fold uses WMMA.
// ---------------------------------------------------------------------------
__global__ void k_main(const float* __restrict__ preds,
                       const float* __restrict__ tgt_box,
                       const int*   __restrict__ tgt_cls,
                       const int*   __restrict__ map,
                       float* __restrict__ partials) {
    float acc = 0.0f;
    const int stride = gridDim.x * blockDim.x;
    for (int c = blockIdx.x * blockDim.x + threadIdx.x; c < NCELLS; c += stride) {
        if (c + stride < NCELLS)
            __builtin_prefetch(&preds[(size_t)(c + stride) * DD + 4], 0, 0);
        const float* p = preds + (size_t)c * DD;
        float pobj = p[4];
        int mv = map[c];
        if (mv == 0) {
            acc += LAMBDA_NOOBJ * pobj * pobj;
        } else {
            int m = mv - 1;
            int b = c / (AA * GH * GW);
            const float* tb = tgt_box + (size_t)(b * MM + m) * 4;
            float gx = tb[0] * GW, gy = tb[1] * GH;
            float bx = gx - floorf(gx), by = gy - floorf(gy);
            float bw = tb[2] * GW,     bh = tb[3] * GH;
            int cls = tgt_cls[b * MM + m];
            float d0 = p[0] - bx, d1 = p[1] - by, d2 = p[2] - bw, d3 = p[3] - bh;
            acc += LAMBDA_BOX * (d0 * d0 + d1 * d1 + d2 * d2 + d3 * d3);
            float dj = pobj - 1.0f;
            acc += LAMBDA_OBJ * dj * dj;
            float cs = 0.0f;
            for (int j = 0; j < CC; ++j) {
                float tt = (j == cls) ? 1.0f : 0.0f;
                float dd = p[5 + j] - tt;
                cs += dd * dd;
            }
            acc += LAMBDA_CLS * cs;
        }
    }
    // block reduction: WMMA wave fold -> LDS -> thread 0
    float wsum = wave_sum_wmma(acc);
    __shared__ float ws[NT2 / 32];
    int lane = threadIdx.x & 31, wid = threadIdx.x >> 5;
    if (lane == 0) ws[wid] = wsum;
    __syncthreads();
    if (threadIdx.x == 0) {
        float bsum = 0.0f;
        #pragma unroll
        for (int i = 0; i < NT2 / 32; ++i) bsum += ws[i];
        partials[blockIdx.x] = bsum;
    }
}

// ---------------------------------------------------------------------------
// K3: single-wave final reduction. C += ones(16x4) x B(4x16) folds 64 block
// partials per WMMA; columns accumulate, rows are identical copies.
// ---------------------------------------------------------------------------
__global__ void k_final(const float* __restrict__ partials, float* __restrict__ out) {
    v2f a; a[0] = 1.0f; a[1] = 1.0f;
    v8f acc = {};
    int lane = threadIdx.x;  // blockDim = 32, one full wave
    for (int base = 0; base < NB2; base += 64) {
        int i0 = base + lane;
        int i1 = base + 32 + lane;
        v2f b;
        b[0] = (i0 < NB2) ? partials[i0] : 0.0f;
        b[1] = (i1 < NB2) ? partials[i1] : 0.0f;
        acc = __builtin_amdgcn_wmma_f32_16x16x4_f32(false, a, false, b,
                                                    (short)0, acc, false, false);
    }
    float s = acc[0] + acc[1] + acc[2] + acc[3] + acc[4] + acc[5] + acc[6] + acc[7];
    s += __shfl_xor(s, 1, 32);
    s += __shfl_xor(s, 2, 32);
    s += __shfl_xor(s, 4, 32);
    s += __shfl_xor(s, 8, 32);   // s = 8 * grand_total on every lane
    if (lane == 0) out[0] = s * 0.125f;
}

// ---------------------------------------------------------------------------
extern "C" void kernel_launch(void* const* d_in, const int* in_sizes, int n_in,
                              void* d_out, int out_size, void* d_ws, size_t ws_size,
                              hipStream_t stream) {
    const float* preds   = (const float*)d_in[0];
    const float* tgt_box = (const float*)d_in[1];
    const float* anchors = (const float*)d_in[2];
    const int*   tgt_cls = (const int*)  d_in[3];

    int*   map      = (int*)d_ws;                       // NCELLS ints
    float* partials = (float*)((char*)d_ws + (size_t)NCELLS * sizeof(int)); // NB2 floats
    float* out      = (float*)d_out;

    k_zero_map<<<(NCELLS + 255) / 256, 256, 0, stream>>>(map);
    k_assign  <<<(BB * MM + 255) / 256, 256, 0, stream>>>(tgt_box, tgt_cls, anchors, map);
    k_main    <<<NB2, NT2, 0, stream>>>(preds, tgt_box, tgt_cls, map, partials);
    k_final   <<<1, 32, 0, stream>>>(partials, out);
}